// PointSampler_6906307412164
// MI455X (gfx1250) — compile-verified
//
#include <hip/hip_runtime.h>
#include <math.h>

#define NN 100000
#define NE 1600000
#define D 64
#define NL 3
#define LDSP 68   // padded LDS row stride (floats) to avoid bank conflicts

typedef __attribute__((ext_vector_type(2))) float v2f;
typedef __attribute__((ext_vector_type(8))) float v8f;

// ---------------- init agg to -inf ----------------
__global__ __launch_bounds__(256) void init_neginf_kernel(float* __restrict__ agg) {
    long long i = (long long)blockIdx.x * 256 + threadIdx.x;
    if (i < (long long)NN * D) agg[i] = -INFINITY;
}

// ---------------- fused edge kernel ----------------
// per wave: 16-edge tile. msg = (x[src]-x[dst]) @ Wp^T  (16x64 = A[16x64] * B[64x64])
// then atomic segment-max into agg[dst].
__global__ __launch_bounds__(256) void devconv_edge_kernel(
    const float* __restrict__ x,
    const long long* __restrict__ src,
    const long long* __restrict__ dst,
    const float* __restrict__ Wp,    // [64][64] row-major: Wp[n][k]
    float* __restrict__ agg)         // [NN][64], pre-initialized to -inf
{
    __shared__ float sW[D * LDSP];
    const int tid = threadIdx.x;
    for (int i = tid; i < D * D; i += 256) {
        int r = i >> 6, c = i & 63;
        sW[r * LDSP + c] = Wp[i];
    }
    __syncthreads();

    const int wave = tid >> 5;
    const int lane = tid & 31;
    const int half = lane >> 4;        // 0: lanes 0-15, 1: lanes 16-31
    const int m    = lane & 15;
    const int koff = half * 2;         // K sub-offset within a K-block of 4

    const long long tileBase = ((long long)blockIdx.x * 8 + wave) * 16;

    // A tile: this lane supplies row M=m, K = 4*kb + koff + {0,1}
    const long long e  = tileBase + m;
    const float* xs = x + src[e] * D;
    const float* xd = x + dst[e] * D;

    v2f a[16];
#pragma unroll
    for (int kb = 0; kb < 16; ++kb) {
        const int c0 = kb * 4 + koff;
        float2 vs = *(const float2*)(xs + c0);
        float2 vd = *(const float2*)(xd + c0);
        a[kb].x = vs.x - vd.x;
        a[kb].y = vs.y - vd.y;
    }

    // destination nodes for the 8 C rows this lane holds (M = r + 8*half)
    int dnode[8];
#pragma unroll
    for (int r = 0; r < 8; ++r)
        dnode[r] = (int)dst[tileBase + r + half * 8];

#pragma unroll
    for (int nt = 0; nt < 4; ++nt) {
        v8f acc = {};
        // B: lane supplies column N = 16*nt + m, K = 4*kb + koff + {0,1}
        const float* wrow = sW + (nt * 16 + m) * LDSP + koff;
#pragma unroll
        for (int kb = 0; kb < 16; ++kb) {
            v2f b;
            b.x = wrow[kb * 4 + 0];
            b.y = wrow[kb * 4 + 1];
            acc = __builtin_amdgcn_wmma_f32_16x16x4_f32(
                false, a[kb], false, b, (short)0, acc, false, false);
        }
        const int ncol = nt * 16 + m;
#pragma unroll
        for (int r = 0; r < 8; ++r) {
            float* p = agg + (long long)dnode[r] * D + ncol;
            __hip_atomic_fetch_max(p, acc[r], __ATOMIC_RELAXED,
                                   __HIP_MEMORY_SCOPE_AGENT);
        }
    }
}

// ---------------- node kernel ----------------
// x_next = clean(agg) @ Wt^T  (per wave: 16-node tile)
__global__ __launch_bounds__(256) void devconv_node_kernel(
    const float* __restrict__ agg,
    const float* __restrict__ Wt,    // [64][64] row-major: Wt[n][k]
    float* __restrict__ xout)        // [NN][64]
{
    __shared__ float sW[D * LDSP];
    const int tid = threadIdx.x;
    for (int i = tid; i < D * D; i += 256) {
        int r = i >> 6, c = i & 63;
        sW[r * LDSP + c] = Wt[i];
    }
    __syncthreads();

    const int wave = tid >> 5;
    const int lane = tid & 31;
    const int half = lane >> 4;
    const int m    = lane & 15;
    const int koff = half * 2;

    const int tile = blockIdx.x * 8 + wave;        // wave-uniform
    if (tile * 16 >= NN) return;                   // uniform guard: EXEC stays full
    const long long tileBase = (long long)tile * 16;

    const float* arow = agg + (tileBase + m) * D;
    v2f a[16];
#pragma unroll
    for (int kb = 0; kb < 16; ++kb) {
        const int c0 = kb * 4 + koff;
        float2 v = *(const float2*)(arow + c0);
        a[kb].x = (v.x == -INFINITY) ? 0.0f : v.x;   // isolated nodes -> 0
        a[kb].y = (v.y == -INFINITY) ? 0.0f : v.y;
    }

#pragma unroll
    for (int nt = 0; nt < 4; ++nt) {
        v8f acc = {};
        const float* wrow = sW + (nt * 16 + m) * LDSP + koff;
#pragma unroll
        for (int kb = 0; kb < 16; ++kb) {
            v2f b;
            b.x = wrow[kb * 4 + 0];
            b.y = wrow[kb * 4 + 1];
            acc = __builtin_amdgcn_wmma_f32_16x16x4_f32(
                false, a[kb], false, b, (short)0, acc, false, false);
        }
        const int ncol = nt * 16 + m;
#pragma unroll
        for (int r = 0; r < 8; ++r) {
            xout[(tileBase + r + half * 8) * D + ncol] = acc[r];
        }
    }
}

// ---------------- scoring head ----------------
__global__ __launch_bounds__(256) void score_kernel(
    const float* __restrict__ x,
    const float* __restrict__ Wout,   // [64]
    const float* __restrict__ bout,   // [1]
    float* __restrict__ out)
{
    int i = blockIdx.x * 256 + threadIdx.x;
    if (i >= NN) return;
    const float4* xv = (const float4*)(x + (long long)i * D);
    const float4* wv = (const float4*)Wout;
    float s = 0.0f;
#pragma unroll
    for (int c = 0; c < 16; ++c) {
        float4 xx = xv[c], ww = wv[c];
        s += xx.x * ww.x + xx.y * ww.y + xx.z * ww.z + xx.w * ww.w;
    }
    s += bout[0];
    out[i] = 1.0f / (1.0f + __expf(-s));
}

extern "C" void kernel_launch(void* const* d_in, const int* in_sizes, int n_in,
                              void* d_out, int out_size, void* d_ws, size_t ws_size,
                              hipStream_t stream) {
    const float*      x      = (const float*)d_in[0];
    const long long*  edges  = (const long long*)d_in[1];   // [2][NE] int64
    const float*      W_phi  = (const float*)d_in[2];        // [3][64][64]
    const float*      W_theta= (const float*)d_in[3];        // [3][64][64]
    const float*      W_out  = (const float*)d_in[4];        // [1][64]
    const float*      b_out  = (const float*)d_in[5];        // [1]
    float*            out    = (float*)d_out;

    const long long* src = edges;
    const long long* dst = edges + NE;

    float* bufA = (float*)d_ws;                       // NN*D floats
    float* bufB = bufA + (size_t)NN * D;
    float* agg  = bufB + (size_t)NN * D;

    const float* cur = x;
    float* nxt = bufA;

    const int initBlocks = (NN * D + 255) / 256;      // 25000
    const int edgeBlocks = NE / (16 * 8);             // 12500 (exact)
    const int nodeBlocks = (NN / 16 + 7) / 8;         // 782 (wave-uniform tail guard)

    for (int l = 0; l < NL; ++l) {
        init_neginf_kernel<<<initBlocks, 256, 0, stream>>>(agg);
        devconv_edge_kernel<<<edgeBlocks, 256, 0, stream>>>(
            cur, src, dst, W_phi + (size_t)l * D * D, agg);
        devconv_node_kernel<<<nodeBlocks, 256, 0, stream>>>(
            agg, W_theta + (size_t)l * D * D, nxt);
        cur = nxt;
        nxt = (nxt == bufA) ? bufB : bufA;
    }
    score_kernel<<<(NN + 255) / 256, 256, 0, stream>>>(cur, W_out, b_out, out);
}